// NeuralGraphHidden_32504312496730
// MI455X (gfx1250) — compile-verified
//
#include <hip/hip_runtime.h>

typedef __attribute__((ext_vector_type(2))) float v2f;
typedef __attribute__((ext_vector_type(4))) float v4f;
typedef __attribute__((ext_vector_type(8))) float v8f;

#define B_     1024
#define A_     128
#define D_     6
#define FA_    64
#define FB_    8
#define FIN_   72            // 64 + 8
#define CONV_  128
#define KTOT_  432           // D_ * FIN_  (degree-stacked K)
#define TILES_ ((B_ * A_) / 16)   // 8192 M-tiles of 16 atoms
#define SROW_  76            // padded S row stride: 16B aligned, bank-spread

__global__ __launch_bounds__(256)
void ngh_fused(const float* __restrict__ atoms,
               const float* __restrict__ bonds,
               const int*   __restrict__ edges,
               const float* __restrict__ Ws,    // (D_, FIN_, CONV_) == stacked (KTOT_, CONV_)
               const float* __restrict__ bs,    // (D_, CONV_)
               float* __restrict__ out) {       // (B_, A_, CONV_)
  // Per-wave 16-column weight slice, quad-interleaved: Wl[wv][k/4][c][k&3]
  // -> each B fragment is ONE conflict-free ds_load_b64 with an immediate offset.
  __shared__ float Wl[8 * KTOT_ * 16];          // 221184 B
  __shared__ float bsL[D_ * CONV_];             // 3072 B (bias in LDS)
  __shared__ float S[16 * SROW_];               // summed (16 x 72), padded
  __shared__ int   edgS[16 * D_];
  __shared__ int   degS[16];

  const int t    = threadIdx.x;
  const int wv   = t >> 5;          // wave id == N-tile id (0..7)
  const int lane = t & 31;
  const int m    = lane & 15;       // C row (A side) / slice column (B side)
  const int hi   = lane >> 4;       // half-wave: K pair {0,1} vs {2,3}
  const int col  = (wv << 4) + m;   // absolute CONV column

  // ---- one-time: stage weight slice (quad-interleaved) + bias into LDS ----
  for (int q = lane; q < KTOT_ * 16; q += 32) {
    int k = q >> 4;                 // stacked K row 0..431
    int c = q & 15;                 // column within slice (coalesced 64B global)
    Wl[wv * (KTOT_ * 16) + (k >> 2) * 64 + c * 4 + (k & 3)] =
        Ws[k * CONV_ + (wv << 4) + c];
  }
  for (int idx = t; idx < D_ * CONV_; idx += 256) bsL[idx] = bs[idx];

  // per-lane B base: column m, half-select hi -> dwords (k&3) = hi*2, hi*2+1
  const float* bptr = Wl + wv * (KTOT_ * 16) + m * 4 + hi * 2;

  for (int T = blockIdx.x; T < TILES_; T += gridDim.x) {
    const int b  = T >> 3;              // batch (A_/16 == 8 tiles per batch)
    const int a0 = (T & 7) << 4;        // first atom of tile

    if (t < 16 * D_) edgS[t] = edges[(b * A_ + a0 + t / D_) * D_ + (t % D_)];
    __syncthreads();

    if (t < 16) {
      int dg = 0;
      #pragma unroll
      for (int d = 0; d < D_; ++d) dg += (edgS[t * D_ + d] >= 0);
      degS[t] = dg;                     // 0..6 ; 6 -> zero output
    }

    // summed = [atoms + sum(neigh) | sum(bonds)]  (16 x 72), b128 everywhere
    for (int idx = t; idx < 16 * 18; idx += 256) {
      int i = idx / 18;
      int q = idx - i * 18;
      int f = q << 2;                   // 0..68, covers 64 atom + 8 bond feats
      int arow = b * A_ + a0 + i;
      v4f v;
      if (q < 16) {
        v = *(const v4f*)(atoms + arow * FA_ + f);
        #pragma unroll
        for (int d = 0; d < D_; ++d) {
          int e = edgS[i * D_ + d];
          if (e >= 0) v += *(const v4f*)(atoms + (b * A_ + e) * FA_ + f);
        }
      } else {
        v4f z = {0.f, 0.f, 0.f, 0.f};
        v = z;
        #pragma unroll
        for (int d = 0; d < D_; ++d)
          v += *(const v4f*)(bonds + (arow * D_ + d) * FB_ + (f - FA_));
      }
      *(v4f*)(S + i * SROW_ + f) = v;
    }
    __syncthreads();

    // ---- degree-stacked GEMM: 6 blocks x 18 K-steps, fully unrolled ----
    const int    degm  = degS[m];
    const float* Srow2 = S + m * SROW_ + hi * 2;

    // A-side fragments loaded ONCE, reused across all 6 degree blocks
    v2f sv[18];
    #pragma unroll
    for (int s = 0; s < 18; ++s) sv[s] = *(const v2f*)(Srow2 + s * 4);

    v8f acc0 = {}, acc1 = {};           // two accs break the WMMA dep chain
    #pragma unroll
    for (int blk = 0; blk < D_; ++blk) {
      const bool hit = (blk == degm);   // one v_cmp per block (6 total)
      const float* bk = bptr + blk * (18 * 64);
      #pragma unroll
      for (int s = 0; s < 18; ++s) {
        v2f zero = {0.f, 0.f};
        v2f af = hit ? sv[s] : zero;            // cndmask, EXEC stays full
        v2f bf = *(const v2f*)(bk + s * 64);    // ds_load_b64, imm offset
        if (s & 1)
          acc1 = __builtin_amdgcn_wmma_f32_16x16x4_f32(
              false, af, false, bf, (short)0, acc1, false, false);
        else
          acc0 = __builtin_amdgcn_wmma_f32_16x16x4_f32(
              false, af, false, bf, (short)0, acc0, false, false);
      }
    }

    // ---- branchless epilogue: bias[min(deg,5)] from LDS, ReLU, deg==6 gate ----
    #pragma unroll
    for (int r = 0; r < 8; ++r) {
      int   mr  = r + hi * 8;                   // C layout: lanes16-31 -> M+8
      int   dg  = degS[mr];
      int   dgc = dg < (D_ - 1) ? dg : (D_ - 1);
      float v   = acc0[r] + acc1[r] + bsL[dgc * CONV_ + col];
      v = fmaxf(v, 0.f);
      v = (dg < D_) ? v : 0.f;                  // cndmask, no branch
      out[(b * A_ + a0 + mr) * CONV_ + col] = v;
    }
    __syncthreads();   // protect S/edgS/degS before next tile's staging
  }
}

extern "C" void kernel_launch(void* const* d_in, const int* in_sizes, int n_in,
                              void* d_out, int out_size, void* d_ws, size_t ws_size,
                              hipStream_t stream) {
  const float* atoms = (const float*)d_in[0];
  const float* bonds = (const float*)d_in[1];
  const int*   edges = (const int*)d_in[2];
  const float* Ws    = (const float*)d_in[3];
  const float* bs    = (const float*)d_in[4];
  float*       out   = (float*)d_out;
  (void)in_sizes; (void)n_in; (void)out_size; (void)d_ws; (void)ws_size;

  dim3 grid(512), block(256);
  hipLaunchKernelGGL(ngh_fused, grid, block, 0, stream,
                     atoms, bonds, edges, Ws, bs, out);
}